// UltraFastGeodesicModel_50964081935193
// MI455X (gfx1250) — compile-verified
//
#include <hip/hip_runtime.h>
#include <hip/hip_bf16.h>

typedef float v2f __attribute__((ext_vector_type(2)));
typedef float v8f __attribute__((ext_vector_type(8)));

__device__ __forceinline__ float fast_tanh(float x) {
#if __has_builtin(__builtin_amdgcn_tanhf)
    return __builtin_amdgcn_tanhf(x);   // gfx1250 v_tanh_f32, 1 trans op
#else
    return tanhf(x);
#endif
}

// D = A x B + 0 with V_WMMA_F32_16X16X4_F32.
//   M (rows of A / D)   = hidden units (rows 8..15 duplicate 0..7)
//   N (cols of B / D)   = 16 samples of this tile
//   K = 4               = {c_target, wavelength, 0, 1(bias)}
// A layout (16x4 f32): lane m (0..15): VGPR0=A[m][0], VGPR1=A[m][1];
//                      lane m+16     : VGPR0=A[m][2], VGPR1=A[m][3]
// B layout (4x16 f32): VGPR0: lanes0-15=B[0][n], lanes16-31=B[2][n];
//                      VGPR1: lanes0-15=B[1][n], lanes16-31=B[3][n]
// C/D layout: lane n: VGPR r = D[r][n]; lane n+16: VGPR r = D[r+8][n]
// Since A rows 8..15 == rows 0..7, every lane ends with d[r] = preact(hidden r,
// sample tileBase + (lane&15)) -> layer 2 is per-lane, no cross-lane reduction.
__global__ __launch_bounds__(256, 1) void geodesic_wmma_kernel(
    const float* __restrict__ c_target,
    const float* __restrict__ wavelength,
    const float* __restrict__ A_source,
    const float* __restrict__ W1,   // [8,3] row-major
    const float* __restrict__ b1,   // [8]
    const float* __restrict__ W2,   // [1,8]
    const float* __restrict__ b2,   // [1]
    float* __restrict__ out, int n)
{
    const int lane   = (int)(threadIdx.x & 31u);
    const int laneLo = lane & 15;
    const bool hiHalf = lane >= 16;
    const int h = lane & 7;          // hidden unit index for A marshalling

    // ---- hoisted weight marshalling (loop-invariant) ----
    v2f Aop;
    Aop.x = hiHalf ? 0.0f   : W1[h * 3 + 0];   // A[m][2]=0     | A[m][0]=W1[h][0]
    Aop.y = hiHalf ? b1[h]  : W1[h * 3 + 1];   // A[m][3]=b1[h] | A[m][1]=W1[h][1]

    float w2v[8];
#pragma unroll
    for (int r = 0; r < 8; ++r) w2v[r] = W2[r];   // wave-uniform -> s_load
    const float bias2 = b2[0];

    const int nTiles = n >> 4;                         // 16 samples per tile
    const int flat   = (int)(blockIdx.x * blockDim.x + threadIdx.x);
    const int waveId = flat >> 5;
    const int nWaves = (int)((gridDim.x * blockDim.x) >> 5);

    for (int t = waveId; t < nTiles; t += nWaves) {    // wave-uniform trip count
        const int s = (t << 4) + laneLo;
        const float c = c_target[s];                   // lanes 16-31 duplicate 0-15
        const float w = wavelength[s];                 // (same cachelines, no extra BW)

        v2f Bop;
        Bop.x = hiHalf ? 0.0f : c;                     // B[2][n]=0 | B[0][n]=c
        Bop.y = hiHalf ? 1.0f : w;                     // B[3][n]=1 | B[1][n]=w

        v8f acc = {};                                  // C = 0 (bias folded into K)
        v8f d = __builtin_amdgcn_wmma_f32_16x16x4_f32(
            /*neg_a=*/false, Aop, /*neg_b=*/false, Bop,
            /*c_mod=*/(short)0, acc, /*reuse_a=*/false, /*reuse_b=*/false);

        float sum = bias2;
#pragma unroll
        for (int r = 0; r < 8; ++r)
            sum = __builtin_fmaf(fast_tanh(d[r]), w2v[r], sum);

        if (!hiHalf)                                   // low half-wave owns the tile
            out[s] = A_source[s] + sum;
    }
}

extern "C" void kernel_launch(void* const* d_in, const int* in_sizes, int n_in,
                              void* d_out, int out_size, void* d_ws, size_t ws_size,
                              hipStream_t stream) {
    // setup_inputs order: c_source(unused), c_target, wavelength, A_source, W1, b1, W2, b2
    const float* c_target   = (const float*)d_in[1];
    const float* wavelength = (const float*)d_in[2];
    const float* A_source   = (const float*)d_in[3];
    const float* W1         = (const float*)d_in[4];
    const float* b1         = (const float*)d_in[5];
    const float* W2         = (const float*)d_in[6];
    const float* b2         = (const float*)d_in[7];
    float* out = (float*)d_out;
    const int n = in_sizes[1];

    dim3 grid(4096), block(256);   // 32768 wave32s grid-striding over n/16 tiles
    hipLaunchKernelGGL(geodesic_wmma_kernel, grid, block, 0, stream,
                       c_target, wavelength, A_source, W1, b1, W2, b2, out, n);
}